// WeightedRuleLayer_73366631350443
// MI455X (gfx1250) — compile-verified
//
#include <hip/hip_runtime.h>
#include <math.h>

// CDNA5 wave32 WMMA fragments
typedef __attribute__((ext_vector_type(2))) float v2f;
typedef __attribute__((ext_vector_type(8))) float v8f;

#define NUM_WEIGHTS 8
#define D_IN 16
#define D_OUT 16

// y[n,o] = tanh( sum_w sum_i table[idx[w,n], i] * W[w,o,i] )
// Tile = 16 neurons. Per weight: A(16x16 gathered) x B(16x16 = W^T) via
// 4 chained V_WMMA_F32_16X16X4_F32 (K=4 each), accumulate over 8 weights.
__global__ __launch_bounds__(256)
void WeightedRuleLayer_wmma_kernel(const float* __restrict__ table,   // [NUM_SOURCE, 16]
                                   const float* __restrict__ weights, // [8, 16, 16]
                                   const int*   __restrict__ gidx,    // [8, num_neurons]
                                   float*       __restrict__ out,     // [num_neurons, 16]
                                   int num_neurons, int num_tiles)
{
    const int lane   = threadIdx.x & 31;
    const int wave   = blockIdx.x * (blockDim.x >> 5) + (threadIdx.x >> 5);
    const int nwaves = gridDim.x * (blockDim.x >> 5);

    const int m     = lane & 15;        // A: row M  /  B,D: column N
    const int khalf = (lane >> 4) * 2;  // upper half-wave handles K+2,K+3

    // ---- Preload all B fragments (weights, 8 KB total) into VGPRs ----
    // B[k][n] = W[o=n][i=k]  (so D[m][n] = sum_k g[m,k] * W[n,k])
    // Fragment layout (4x16, 2 VGPRs): lane l -> col n=l&15;
    //   vgpr0 = B[khalf+0][n], vgpr1 = B[khalf+1][n]
    v2f bfrag[NUM_WEIGHTS][4];
#pragma unroll
    for (int w = 0; w < NUM_WEIGHTS; ++w) {
        const float* Wrow = weights + w * (D_OUT * D_IN) + m * D_IN; // row o = n
#pragma unroll
        for (int kb = 0; kb < 4; ++kb) {
            const int k = kb * 4 + khalf;
            v2f b;
            b.x = Wrow[k];       // contiguous 8B -> global_load_b64
            b.y = Wrow[k + 1];
            bfrag[w][kb] = b;
        }
    }

    for (int t = wave; t < num_tiles; t += nwaves) {
        const int nbase = t * 16;

        // Indices for this tile: lane's row m, all 8 weights (L2-friendly 64B reads)
        int idxw[NUM_WEIGHTS];
#pragma unroll
        for (int w = 0; w < NUM_WEIGHTS; ++w)
            idxw[w] = gidx[(size_t)w * num_neurons + nbase + m];

        v8f acc = {0.f, 0.f, 0.f, 0.f, 0.f, 0.f, 0.f, 0.f};

#pragma unroll
        for (int w = 0; w < NUM_WEIGHTS; ++w) {
            const float* src = table + (size_t)idxw[w] * D_IN; // gathered row (64B, L2 hit)
#pragma unroll
            for (int kb = 0; kb < 4; ++kb) {
                const int k = kb * 4 + khalf;
                v2f a;
                a.x = src[k];       // 8B per lane; lane pair (l, l+16) + 4 kb
                a.y = src[k + 1];   // covers each gathered row exactly once
                // D = A x B + C  (8 args: neg_a, A, neg_b, B, c_mod, C, reuse_a, reuse_b)
                acc = __builtin_amdgcn_wmma_f32_16x16x4_f32(
                        false, a, false, bfrag[w][kb], (short)0, acc, false, false);
            }
        }

        // D layout: vgpr v -> row M = v + (l>>4)*8, col N = m. Rows coalesce to 64B.
        float* orow = out + (size_t)nbase * D_OUT;
#pragma unroll
        for (int v = 0; v < 8; ++v) {
            const int row = v + (lane >> 4) * 8;
            orow[row * D_OUT + m] = tanhf(acc[v]);
        }
    }
}

extern "C" void kernel_launch(void* const* d_in, const int* in_sizes, int n_in,
                              void* d_out, int out_size, void* d_ws, size_t ws_size,
                              hipStream_t stream) {
    const float* table   = (const float*)d_in[0];  // layer_values [1M,16] f32
    const float* weights = (const float*)d_in[1];  // [8,16,16] f32
    const int*   gidx    = (const int*)d_in[2];    // [8, num_neurons]
    float*       out     = (float*)d_out;          // [num_neurons, 16] f32

    const int num_neurons = in_sizes[2] / NUM_WEIGHTS;   // 500000
    const int num_tiles   = num_neurons / 16;            // 31250 (exact)

    // 256 threads = 8 wave32 per block; 8192 waves grid-stride over 31250 tiles.
    dim3 block(256);
    dim3 grid(1024);
    WeightedRuleLayer_wmma_kernel<<<grid, block, 0, stream>>>(
        table, weights, gidx, out, num_neurons, num_tiles);
}